// WeightedRankPairwiseLoss2_19971597927132
// MI455X (gfx1250) — compile-verified
//
#include <hip/hip_runtime.h>
#include <hip/hip_bf16.h>

#define TPB 256
#define DEPTH 4   // async pipeline depth (LDS buffers / tiles in flight)

typedef int v4i __attribute__((ext_vector_type(4)));

// ---------------- CDNA5 async-to-LDS helpers (gfx1250) ----------------

__device__ __forceinline__ void async_copy_f4(const float4* g, float4* l) {
#if defined(__HIP_DEVICE_COMPILE__)
#if __has_builtin(__builtin_amdgcn_global_load_async_to_lds_b128)
  __builtin_amdgcn_global_load_async_to_lds_b128(
      (__attribute__((address_space(1))) v4i*)(v4i*)(void*)const_cast<float4*>(g),
      (__attribute__((address_space(3))) v4i*)(v4i*)(void*)l,
      /*imm offset=*/0, /*cpol=*/1 /* TH_NT: read-once stream, evict-first */);
#else
  unsigned loff = (unsigned)(unsigned long long)
      (__attribute__((address_space(3))) char*)(void*)l;
  unsigned long long ga = (unsigned long long)g;
  asm volatile("global_load_async_to_lds_b128 %0, %1, off"
               :: "v"(loff), "v"(ga) : "memory");
#endif
#else
  (void)g; (void)l;
#endif
}

template <int N>
__device__ __forceinline__ void wait_async_le() {
#if defined(__HIP_DEVICE_COMPILE__)
#if __has_builtin(__builtin_amdgcn_s_wait_asynccnt)
  __builtin_amdgcn_s_wait_asynccnt(N);
#else
  asm volatile("s_wait_asynccnt %0" :: "n"(N) : "memory");
#endif
#endif
}

__device__ __forceinline__ void wait_ds0() {
#if defined(__HIP_DEVICE_COMPILE__)
#if __has_builtin(__builtin_amdgcn_s_wait_dscnt)
  __builtin_amdgcn_s_wait_dscnt(0);
#else
  asm volatile("s_wait_dscnt 0x0" ::: "memory");
#endif
#endif
}

// ---------------- Kernel 1: harmonic-number prefix table ----------------
// harm[k] = H(k) = sum_{i=1..k} 1/i ; harm[0] = 0. Single 1024-thread block.

__global__ __launch_bounds__(1024) void harm_scan(float* __restrict__ harm, int C) {
  __shared__ float part[1024];
  const int t = threadIdx.x;
  const int chunk = (C + 1023) / 1024;
  const int s = t * chunk;
  int e = s + chunk; if (e > C) e = C;
  const int s1 = (s > 1) ? s : 1;          // skip i==0 (term 0)

  float sum = 0.0f;
  for (int i = s1; i < e; ++i) sum += 1.0f / (float)i;
  part[t] = sum;
  __syncthreads();

  // inclusive Hillis-Steele scan over 1024 partials
  for (int off = 1; off < 1024; off <<= 1) {
    float add = (t >= off) ? part[t - off] : 0.0f;
    float v = part[t];
    __syncthreads();
    part[t] = v + add;
    __syncthreads();
  }

  float run = (t > 0) ? part[t - 1] : 0.0f;  // exclusive prefix of my chunk
  if (s == 0 && e > 0) harm[0] = 0.0f;
  for (int i = s1; i < e; ++i) {
    run += 1.0f / (float)i;
    harm[i] = run;
  }
}

// ---------------- Kernel 2: per-row rank + hinge via async pipeline ----------------

__global__ __launch_bounds__(TPB) void wrp_row_kernel(
    const float* __restrict__ scores, const int* __restrict__ targets,
    const float* __restrict__ harm, float* __restrict__ contrib, int C) {
  __shared__ float4 tile[DEPTH][TPB];      // 16 KB quad buffer
  __shared__ float red_h[TPB / 32];
  __shared__ int   red_c[TPB / 32];

  const int row  = blockIdx.x;
  const int tid  = threadIdx.x;
  const int nvec = C >> 2;                 // float4s per row (C % 4 == 0)
  const int NT   = (nvec + TPB - 1) / TPB; // tiles per row

  const float4* rowp = (const float4*)(scores + (long long)row * C);
  const int   tgt = targets[row];
  const float gt  = scores[(long long)row * C + tgt];
  const float c1  = 1.0f - gt;             // hinge term = s + (1 - gt)

  // prologue: up to DEPTH tiles in flight (clamped -> uniform async count/wave)
  for (int p = 0; p < DEPTH && p < NT; ++p) {
    int g = p * TPB + tid; if (g > nvec - 1) g = nvec - 1;
    async_copy_f4(&rowp[g], &tile[p & (DEPTH - 1)][tid]);
  }

  float hinge = 0.0f;
  int   cnt   = 0;
  for (int i = 0; i < NT; ++i) {
    const int ahead = NT - 1 - i;          // async issues newer than tile i
    if (ahead >= 3)      wait_async_le<3>();
    else if (ahead == 2) wait_async_le<2>();
    else if (ahead == 1) wait_async_le<1>();
    else                 wait_async_le<0>();

    float4 v = tile[i & (DEPTH - 1)][tid];
    const int vidx = i * TPB + tid;
    if (vidx < nvec) {
      cnt += (int)(v.x > gt) + (int)(v.y > gt) + (int)(v.z > gt) + (int)(v.w > gt);
      hinge += fmaxf(v.x + c1, 0.0f) + fmaxf(v.y + c1, 0.0f)
             + fmaxf(v.z + c1, 0.0f) + fmaxf(v.w + c1, 0.0f);
    }
    wait_ds0();                 // retire LDS reads before WAR reuse of buffer
    const int nx = i + DEPTH;
    if (nx < NT) {
      int g = nx * TPB + tid; if (g > nvec - 1) g = nvec - 1;
      async_copy_f4(&rowp[g], &tile[nx & (DEPTH - 1)][tid]);
    }
  }

  // wave32 reduction, then cross-wave via LDS
  for (int o = 16; o > 0; o >>= 1) {
    hinge += __shfl_xor(hinge, o, 32);
    cnt   += __shfl_xor(cnt, o, 32);
  }
  const int wave = tid >> 5;
  if ((tid & 31) == 0) { red_h[wave] = hinge; red_c[wave] = cnt; }
  __syncthreads();
  if (tid == 0) {
    float h = 0.0f; int c = 0;
    for (int w = 0; w < TPB / 32; ++w) { h += red_h[w]; c += red_c[w]; }
    h -= 1.0f;                                // remove target's relu(margin)=1 term
    float wgt = 0.0f;
    if (c > 0) wgt = harm[c] / (float)c;      // H(rank)/rank, rank==0 -> 0
    contrib[row] = wgt * h;
  }
}

// ---------------- Kernel 3: deterministic mean ----------------

__global__ __launch_bounds__(256) void final_mean(const float* __restrict__ contrib,
                                                  float* __restrict__ out, int B) {
  __shared__ float sp[8];
  const int tid = threadIdx.x;
  float s = 0.0f;
  for (int i = tid; i < B; i += 256) s += contrib[i];
  for (int o = 16; o > 0; o >>= 1) s += __shfl_xor(s, o, 32);
  if ((tid & 31) == 0) sp[tid >> 5] = s;
  __syncthreads();
  if (tid == 0) {
    float t = 0.0f;
    for (int w = 0; w < 8; ++w) t += sp[w];
    out[0] = t / (float)B;
  }
}

// ---------------- Launch ----------------

extern "C" void kernel_launch(void* const* d_in, const int* in_sizes, int n_in,
                              void* d_out, int out_size, void* d_ws, size_t ws_size,
                              hipStream_t stream) {
  const float* scores  = (const float*)d_in[0];
  const int*   targets = (const int*)d_in[1];
  const int B = in_sizes[1];
  const int C = in_sizes[0] / B;   // 32000, multiple of 4

  float* harm    = (float*)d_ws;   // C floats
  float* contrib = harm + C;       // B floats

  harm_scan<<<1, 1024, 0, stream>>>(harm, C);
  wrp_row_kernel<<<B, TPB, 0, stream>>>(scores, targets, harm, contrib, C);
  final_mean<<<1, 256, 0, stream>>>(contrib, (float*)d_out, B);
}